// predictor_interp2d_25890062860484
// MI455X (gfx1250) — compile-verified
//
#include <hip/hip_runtime.h>

typedef __attribute__((ext_vector_type(2))) float v2f;
typedef __attribute__((ext_vector_type(8))) float v8f;

#define BB 8
#define CC 4
#define NN 1000
#define HH 128
#define WW 128
#define GG (HH * WW)
#define NT 64            // candidate tiles of 16 (padded from ceil(1000/16)=63)
#define NPAD (NT * 16)   // 1024 padded candidates

// One wave (32 lanes) handles one tile of 16 grid points for one batch.
// Block = 256 threads = 8 waves = 8 grid tiles. 128 blocks per batch.
__global__ __launch_bounds__(256) void nn_interp_wmma_kernel(
    const float* __restrict__ R_pc,    // [B, C, N]
    const float* __restrict__ XY_pc,   // [B, 2, N]
    const float* __restrict__ XY_grd,  // [H, W, 2]
    float* __restrict__ out)           // [B, C, H, W]
{
  // Two interleaved fragment tables so every lane issues ONE uniform
  // ds_load_b64 per candidate tile (no divergence in the hot loop):
  //   lanes 0-15  read s_xy[m] = {px, py}        (A rows K=0,1)
  //   lanes 16-31 read s_nh[m] = {|p|^2, 1.0}    (A rows K=2,3)
  __shared__ float s_xy[NPAD * 2];
  __shared__ float s_nh[NPAD * 2];

  const int batch    = blockIdx.x >> 7;    // / 128
  const int blk_in_b = blockIdx.x & 127;
  const int tid      = threadIdx.x;

  // ---- stage candidate data for this batch into LDS ----
  const float* px = XY_pc + batch * 2 * NN;
  const float* py = px + NN;
  for (int i = tid; i < NPAD; i += 256) {
    float x = 0.0f, y = 0.0f, nrm = 1e38f;  // sentinel: never the argmin (real d2 <= 8)
    if (i < NN) { x = px[i]; y = py[i]; nrm = x * x + y * y; }
    s_xy[2 * i]     = x;
    s_xy[2 * i + 1] = y;
    s_nh[2 * i]     = nrm;
    s_nh[2 * i + 1] = 1.0f;
  }
  __syncthreads();

  const int  wave = tid >> 5;
  const int  lane = tid & 31;
  const int  col  = lane & 15;
  const bool hi   = lane >= 16;

  const int gtile = blk_in_b * 8 + wave;   // 0..1023
  const int g     = gtile * 16 + col;      // this lane's grid point (column of D)
  const float gx  = XY_grd[2 * g];
  const float gy  = XY_grd[2 * g + 1];

  // B fragment (4x16, fixed over the whole loop):
  //   rows K0..K3 of column n are [-2gx, -2gy, 1, gx^2+gy^2]
  //   lanes 0-15 hold K=0,1 ; lanes 16-31 hold K=2,3
  v2f bfrag;
  if (!hi) { bfrag.x = -2.0f * gx; bfrag.y = -2.0f * gy; }
  else     { bfrag.x = 1.0f;       bfrag.y = gx * gx + gy * gy; }

  // Per-lane A-fragment base pointer: one address select, then uniform loads.
  const float* aptr = (hi ? s_nh : s_xy) + 2 * col;

  float minv = 3.402823466e38f;
  int   mini = 0;

#pragma unroll 2
  for (int t = 0; t < NT; ++t) {
    // A fragment (16x4): row m = [px, py, |p|^2, 1]
    const v2f afrag = *(const v2f*)(aptr + 32 * t);

    v8f d = {};
    // D[m][n] = -2 p.g + |p|^2 + |g|^2 = squared distance, exactly.
    d = __builtin_amdgcn_wmma_f32_16x16x4_f32(
        /*neg_a=*/false, afrag, /*neg_b=*/false, bfrag,
        /*c_mod=*/(short)0, d, /*reuse_a=*/false, /*reuse_b=*/false);

    // D layout: this lane holds candidates M = base..base+7 for its column.
    const int base = t * 16 + (hi ? 8 : 0);
#pragma unroll
    for (int r = 0; r < 8; ++r) {
      float v = d[r];
      if (v < minv) { minv = v; mini = base + r; }
    }
  }

  // Merge the two candidate halves held by the lane pair (l, l^16).
  float ov = __shfl_xor(minv, 16, 32);
  int   oi = __shfl_xor(mini, 16, 32);
  if (ov < minv || (ov == minv && oi < mini)) { minv = ov; mini = oi; }

  // Lanes 0-15 gather the 4 channel values and write out.
  if (!hi) {
    const float* R = R_pc + batch * CC * NN;
    float*       o = out + batch * CC * GG + g;
#pragma unroll
    for (int c = 0; c < CC; ++c) o[c * GG] = R[c * NN + mini];
  }
}

extern "C" void kernel_launch(void* const* d_in, const int* in_sizes, int n_in,
                              void* d_out, int out_size, void* d_ws, size_t ws_size,
                              hipStream_t stream) {
  const float* R_pc   = (const float*)d_in[0];
  const float* XY_pc  = (const float*)d_in[1];
  const float* XY_grd = (const float*)d_in[2];
  float*       out    = (float*)d_out;

  // 8 batches * 128 blocks per batch; each block = 8 waves = 8 grid tiles of 16.
  nn_interp_wmma_kernel<<<dim3(BB * 128), dim3(256), 0, stream>>>(
      R_pc, XY_pc, XY_grd, out);
}